// GIN_MLP_2353642078898
// MI455X (gfx1250) — compile-verified
//
#include <hip/hip_runtime.h>
#include <hip/hip_bf16.h>

typedef __attribute__((ext_vector_type(2))) float v2f;
typedef __attribute__((ext_vector_type(8))) float v8f;

#define N_NODES 100000
#define N_EDGES 1600000
#define HDIM    128
#define BQ      4096
#define CTXN    20

// ---------------- degree: deg[dst[e]] += 1 ----------------
__global__ void deg_kernel(const int* __restrict__ dst, float* __restrict__ deg, int E) {
    int e = blockIdx.x * blockDim.x + threadIdx.x;
    if (e < E) atomicAdd(&deg[dst[e]], 1.0f);
}

// ---------------- scatter: agg[dst[e]][c] += w[e] * x[src[e]][c] ----------------
// one wave per edge, 4 channels per lane (float4 gather + 4 atomics)
__global__ void scatter_kernel(const int* __restrict__ src, const int* __restrict__ dst,
                               const float* __restrict__ w, const float* __restrict__ x,
                               float* __restrict__ agg, int E) {
    long long tid = (long long)blockIdx.x * blockDim.x + threadIdx.x;
    long long e = tid >> 5;
    if (e >= E) return;
    int c = ((int)tid & 31) << 2;
    float wv = w[e];
    int s = src[e], d = dst[e];
    const float4 v = *(const float4*)&x[(long long)s * HDIM + c];
    float* p = &agg[(long long)d * HDIM + c];
    atomicAdd(p + 0, wv * v.x);
    atomicAdd(p + 1, wv * v.y);
    atomicAdd(p + 2, wv * v.z);
    atomicAdd(p + 3, wv * v.w);
}

// ---------------- combine: t = (1+eps)*x + agg/max(deg,1)   (in place over agg) ----------------
__global__ void combine_kernel(const float* __restrict__ x, float* __restrict__ agg,
                               const float* __restrict__ deg, const float* __restrict__ eps,
                               int eidx, long long total) {
    long long idx = (long long)blockIdx.x * blockDim.x + threadIdx.x;
    if (idx >= total) return;
    long long node = idx >> 7;
    float d = fmaxf(deg[node], 1.0f);
    agg[idx] = (1.0f + eps[eidx]) * x[idx] + agg[idx] / d;
}

// ---------------- specialized WMMA fp32 GEMM, K=128, W register-resident ----------------
// blockDim.x = (Nout/16)*32; each wave owns one 16-col tile and preloads all 32 W
// fragments (64 VGPRs) once, then grid-strides over 16-row tiles of A.
// Steady state: 32 x v_wmma per 32 x global_load_b64 (A only) + 8 stores.
__global__ void __launch_bounds__(256)
gemm_wmma_k128_kernel(const float* __restrict__ A, const float* __restrict__ W,
                      const float* __restrict__ bias, float* __restrict__ out,
                      int M, int Nout, int relu) {
    const int lane = threadIdx.x & 31;
    const int wave = threadIdx.x >> 5;          // == column tile index
    const int col0 = wave << 4;
    const int ln15 = lane & 15;
    const int kb   = (lane >> 4) << 1;          // K-pair select: 0 or 2

    // preload W fragments for this column tile (loop-invariant)
    v2f bf[32];
    {
        const float* wp = W + (long long)(col0 + ln15) * 128 + kb;
        #pragma unroll
        for (int kk = 0; kk < 32; ++kk) {
            bf[kk].x = wp[kk * 4];
            bf[kk].y = wp[kk * 4 + 1];
        }
    }
    const float bv = bias[col0 + ln15];
    const int tiles_m = M >> 4;

    for (int rt = blockIdx.x; rt < tiles_m; rt += gridDim.x) {   // wave-uniform loop
        const long long arow = (long long)((rt << 4) + ln15) * 128 + kb;
        v8f acc = {bv, bv, bv, bv, bv, bv, bv, bv};
        #pragma unroll
        for (int kk = 0; kk < 32; ++kk) {
            v2f a;
            a.x = A[arow + kk * 4];
            a.y = A[arow + kk * 4 + 1];
            acc = __builtin_amdgcn_wmma_f32_16x16x4_f32(
                false, a, false, bf[kk], (short)0, acc, false, false);
        }
        const int orow = (rt << 4) + ((lane >> 4) << 3);  // lanes 16-31 hold M = r+8
        #pragma unroll
        for (int r = 0; r < 8; ++r) {
            float v = acc[r];
            if (relu) v = fmaxf(v, 0.0f);
            out[(long long)(orow + r) * Nout + col0 + ln15] = v;
        }
    }
}

// ---------------- generic WMMA fp32 GEMM (runtime K) — used for the K=384 layer ----------------
__global__ void __launch_bounds__(256)
gemm_wmma_kernel(const float* __restrict__ A, const float* __restrict__ W,
                 const float* __restrict__ bias, float* __restrict__ out,
                 int M, int K, int Nout, int relu) {
    const int lane = threadIdx.x & 31;
    const int wave = threadIdx.x >> 5;
    const int tiles_n = Nout >> 4;
    const int tiles_m = M >> 4;
    const int tile = blockIdx.x * (blockDim.x >> 5) + wave;
    if (tile >= tiles_m * tiles_n) return;          // wave-uniform: EXEC stays all-ones
    const int row0 = (tile / tiles_n) << 4;
    const int col0 = (tile % tiles_n) << 4;
    const int ln15 = lane & 15;
    const int kb   = (lane >> 4) << 1;

    const long long arow = (long long)(row0 + ln15) * K;
    const long long wrow = (long long)(col0 + ln15) * K;

    const float bv = bias[col0 + ln15];
    v8f acc = {bv, bv, bv, bv, bv, bv, bv, bv};

    for (int k = 0; k < K; k += 4) {
        v2f a, b;
        a.x = A[arow + k + kb];
        a.y = A[arow + k + kb + 1];
        b.x = W[wrow + k + kb];
        b.y = W[wrow + k + kb + 1];
        acc = __builtin_amdgcn_wmma_f32_16x16x4_f32(
            false, a, false, b, (short)0, acc, false, false);
    }

    const int orow = row0 + ((lane >> 4) << 3);
    #pragma unroll
    for (int r = 0; r < 8; ++r) {
        float v = acc[r];
        if (relu) v = fmaxf(v, 0.0f);
        out[(long long)(orow + r) * Nout + col0 + ln15] = v;
    }
}

// ---------------- gather: y[b] = [x[i0], x[i1], mean_{ctx>0} x[ic]]  (x row 0 treated as 0) ----
__global__ void gather_kernel(const long long* __restrict__ idx, const float* __restrict__ x,
                              float* __restrict__ y) {
    int tid = blockIdx.x * blockDim.x + threadIdx.x;
    if (tid >= BQ * HDIM) return;
    int b = tid >> 7;
    int k = tid & (HDIM - 1);
    const long long* row = idx + (long long)b * (3 + CTXN);
    long long i0 = row[0], i1 = row[1];
    float* yr = y + (long long)b * (3 * HDIM);
    yr[k]            = (i0 == 0) ? 0.0f : x[i0 * HDIM + k];
    yr[HDIM + k]     = (i1 == 0) ? 0.0f : x[i1 * HDIM + k];
    float s = 0.0f;
    int cnt = 0;
    #pragma unroll
    for (int j = 0; j < CTXN; ++j) {
        long long ic = row[3 + j];
        if (ic > 0) { s += x[ic * HDIM + k]; ++cnt; }
    }
    float norm = (float)(cnt > 0 ? cnt : 1);
    yr[2 * HDIM + k] = s / norm;
}

// ---------------- final: out[b] = dot(h2[b], Wc3) + bc3 ----------------
__global__ void final_kernel(const float* __restrict__ h2, const float* __restrict__ Wc3,
                             const float* __restrict__ bc3, float* __restrict__ out) {
    int b = blockIdx.x * blockDim.x + threadIdx.x;
    if (b >= BQ) return;
    float s = bc3[0];
    #pragma unroll
    for (int k = 0; k < 64; ++k) s += h2[(long long)b * 64 + k] * Wc3[k];
    out[b] = s;
}

static inline int gemm_blocks(int M, int Nout) {
    int tiles = (M >> 4) * (Nout >> 4);
    return (tiles + 7) / 8;  // 8 waves (256 threads) per block
}

extern "C" void kernel_launch(void* const* d_in, const int* in_sizes, int n_in,
                              void* d_out, int out_size, void* d_ws, size_t ws_size,
                              hipStream_t stream) {
    const long long* indices = (const long long*)d_in[0];   // int64 per reference
    const int*   src   = (const int*)d_in[1];
    const int*   dst   = (const int*)d_in[2];
    const float* w     = (const float*)d_in[3];
    const float* ndata = (const float*)d_in[4];
    const float* W1    = (const float*)d_in[5];
    const float* b1    = (const float*)d_in[6];
    const float* W2    = (const float*)d_in[7];
    const float* b2    = (const float*)d_in[8];
    const float* eps   = (const float*)d_in[9];
    const float* Wc1   = (const float*)d_in[10];
    const float* bc1   = (const float*)d_in[11];
    const float* Wc2   = (const float*)d_in[12];
    const float* bc2   = (const float*)d_in[13];
    const float* Wc3   = (const float*)d_in[14];
    const float* bc3   = (const float*)d_in[15];
    float* out = (float*)d_out;

    // workspace carve-up (256B aligned)
    char* ws = (char*)d_ws;
    size_t off = 0;
    auto carve = [&](size_t nfloats) -> float* {
        float* p = (float*)(ws + off);
        off += ((nfloats * sizeof(float) + 255) / 256) * 256;
        return p;
    };
    float* deg = carve(N_NODES);
    float* agg = carve((size_t)N_NODES * HDIM);   // scatter target, then combined "t"
    float* xb  = carve((size_t)N_NODES * HDIM);   // x1, later overwritten by x2
    float* y   = carve((size_t)BQ * 3 * HDIM);
    float* h1  = carve((size_t)BQ * HDIM);
    float* h2  = carve((size_t)BQ * 64);
    (void)ws_size; (void)in_sizes; (void)n_in; (void)out_size;

    const long long NH = (long long)N_NODES * HDIM;
    const int node_tiles = N_NODES / 16;                       // 6250
    const int gemm_grid  = node_tiles < 1024 ? node_tiles : 1024;

    hipMemsetAsync(deg, 0, N_NODES * sizeof(float), stream);
    hipMemsetAsync(agg, 0, NH * sizeof(float), stream);

    // degree
    deg_kernel<<<(N_EDGES + 255) / 256, 256, 0, stream>>>(dst, deg, N_EDGES);

    // ---- layer 1 ----
    {
        long long t = (long long)N_EDGES * 32;
        scatter_kernel<<<(int)((t + 255) / 256), 256, 0, stream>>>(src, dst, w, ndata, agg, N_EDGES);
    }
    combine_kernel<<<(int)((NH + 255) / 256), 256, 0, stream>>>(ndata, agg, deg, eps, 0, NH);
    gemm_wmma_k128_kernel<<<gemm_grid, 256, 0, stream>>>(
        agg, W1, b1, xb, N_NODES, HDIM, /*relu=*/1);

    // ---- layer 2 ----
    hipMemsetAsync(agg, 0, NH * sizeof(float), stream);
    {
        long long t = (long long)N_EDGES * 32;
        scatter_kernel<<<(int)((t + 255) / 256), 256, 0, stream>>>(src, dst, w, xb, agg, N_EDGES);
    }
    combine_kernel<<<(int)((NH + 255) / 256), 256, 0, stream>>>(xb, agg, deg, eps, 1, NH);
    gemm_wmma_k128_kernel<<<gemm_grid, 256, 0, stream>>>(
        agg, W2, b2, xb, N_NODES, HDIM, /*relu=*/0);           // xb now holds x2

    // ---- gather + concat (x2[0] treated as zero inside) ----
    gather_kernel<<<(BQ * HDIM + 255) / 256, 256, 0, stream>>>(indices, xb, y);

    // ---- query MLP ----
    gemm_wmma_kernel<<<gemm_blocks(BQ, HDIM), 256, 0, stream>>>(
        y, Wc1, bc1, h1, BQ, 3 * HDIM, HDIM, /*relu=*/1);      // K=384 generic path
    gemm_wmma_k128_kernel<<<BQ / 16, 128, 0, stream>>>(
        h1, Wc2, bc2, h2, BQ, 64, /*relu=*/1);                 // Nout=64 -> 4 waves/block
    final_kernel<<<(BQ + 255) / 256, 256, 0, stream>>>(h2, Wc3, bc3, out);
}